// SinkhornSolver_11287174053967
// MI455X (gfx1250) — compile-verified
//
#include <hip/hip_runtime.h>
#include <math.h>

#define BATCH 8
#define N 2048
#define D 256
#define EPS 0.1f
#define INV_EPS 10.0f
#define THRESH 0.1f

typedef float v2f __attribute__((ext_vector_type(2)));
typedef float v8f __attribute__((ext_vector_type(8)));

__device__ __forceinline__ void lse_step(float& m, float& s, float k) {
    float nm = fmaxf(m, k);
    s = s * __expf(m - nm) + __expf(k - nm);
    m = nm;
}
__device__ __forceinline__ void lse_merge(float& m, float& s, float m2, float s2) {
    float nm = fmaxf(m, m2);
    s = s * __expf(m - nm) + s2 * __expf(m2 - nm);
    m = nm;
}

// ---------------------------------------------------------------- init
__global__ __launch_bounds__(256) void init_kernel(float* u0, float* v0, int* done) {
    int idx = blockIdx.x * 256 + threadIdx.x;     // 64 blocks -> 16384 threads
    u0[idx] = 0.0f;
    v0[idx] = 0.0f;
    if (idx == 0) *done = 0;
}

// ---------------------------------------------------------------- row norms |x_i|^2, |y_j|^2
__global__ __launch_bounds__(256) void norms_kernel(const float* __restrict__ X,
                                                    const float* __restrict__ Y,
                                                    float* __restrict__ x2,
                                                    float* __restrict__ y2) {
    int idx = blockIdx.x * 256 + threadIdx.x;     // 0..16383 == b*N+i
    const float4* xr = (const float4*)(X + (size_t)idx * D);
    const float4* yr = (const float4*)(Y + (size_t)idx * D);
    float sx = 0.0f, sy = 0.0f;
#pragma unroll 4
    for (int d = 0; d < D / 4; ++d) {
        float4 a = xr[d];
        sx += a.x * a.x + a.y * a.y + a.z * a.z + a.w * a.w;
        float4 b = yr[d];
        sy += b.x * b.x + b.y * b.y + b.z * b.z + b.w * b.w;
    }
    x2[idx] = sx;
    y2[idx] = sy;
}

// ---------------------------------------------------------------- WMMA cost matrix
// C[b,i,j] = x2[b,i] + y2[b,j] - 2 * sum_d X[b,i,d]*Y[b,j,d]
// One wave -> 32x32 output tile via 2x2 grid of v_wmma_f32_16x16x4_f32.
__global__ __launch_bounds__(256) void gemm_cost_kernel(const float* __restrict__ X,
                                                        const float* __restrict__ Y,
                                                        const float* __restrict__ x2,
                                                        const float* __restrict__ y2,
                                                        float* __restrict__ C) {
    const int lane = threadIdx.x;        // 0..31
    const int half = lane >> 4;          // 0/1 -> K pair select
    const int l16  = lane & 15;

    int tile = blockIdx.x * 8 + threadIdx.y;   // 0..32767
    int b  = tile >> 12;                       // / 4096 tiles per batch
    int r  = tile & 4095;
    int m0 = (r >> 6) << 5;                    // 64 tiles per row dim
    int n0 = (r & 63) << 5;

    const float* Xb = X + (size_t)b * N * D;
    const float* Yb = Y + (size_t)b * N * D;

    // A layout (16x4 f32): lane<16 holds M=l16, K = 2*half + {0,1} across 2 VGPRs
    // B layout (4x16 f32): lane<16 holds N=l16, K = 2*half + {0,1} across 2 VGPRs
    const float* xrow0 = Xb + (size_t)(m0 + l16)      * D + 2 * half;
    const float* xrow1 = Xb + (size_t)(m0 + 16 + l16) * D + 2 * half;
    const float* yrow0 = Yb + (size_t)(n0 + l16)      * D + 2 * half;
    const float* yrow1 = Yb + (size_t)(n0 + 16 + l16) * D + 2 * half;

    v8f acc00 = {}, acc01 = {}, acc10 = {}, acc11 = {};

#pragma unroll 4
    for (int k = 0; k < D; k += 4) {
        v2f a0 = *(const v2f*)(xrow0 + k);
        v2f a1 = *(const v2f*)(xrow1 + k);
        v2f b0 = *(const v2f*)(yrow0 + k);
        v2f b1 = *(const v2f*)(yrow1 + k);
        acc00 = __builtin_amdgcn_wmma_f32_16x16x4_f32(false, a0, false, b0, (short)0, acc00, false, false);
        acc01 = __builtin_amdgcn_wmma_f32_16x16x4_f32(false, a0, false, b1, (short)0, acc01, false, false);
        acc10 = __builtin_amdgcn_wmma_f32_16x16x4_f32(false, a1, false, b0, (short)0, acc10, false, false);
        acc11 = __builtin_amdgcn_wmma_f32_16x16x4_f32(false, a1, false, b1, (short)0, acc11, false, false);
    }

    // D layout: VGPR r, lane<16 -> (M = 8*half + r, N = l16)
    const float* x2b = x2 + b * N;
    const float* y2b = y2 + b * N;
    float* Cb = C + (size_t)b * N * N;
    int col0 = n0 + l16, col1 = n0 + 16 + l16;
    float y2c0 = y2b[col0], y2c1 = y2b[col1];
#pragma unroll
    for (int rr = 0; rr < 8; ++rr) {
        int row0 = m0 + 8 * half + rr;
        int row1 = row0 + 16;
        float xr0 = x2b[row0], xr1 = x2b[row1];
        Cb[(size_t)row0 * N + col0] = xr0 + y2c0 - 2.0f * acc00[rr];
        Cb[(size_t)row0 * N + col1] = xr0 + y2c1 - 2.0f * acc01[rr];
        Cb[(size_t)row1 * N + col0] = xr1 + y2c0 - 2.0f * acc10[rr];
        Cb[(size_t)row1 * N + col1] = xr1 + y2c1 - 2.0f * acc11[rr];
    }
}

// ---------------------------------------------------------------- u update: LSE over i (axis=1)
// u_next[b,t] = EPS*(log_ab - LSE_i K[b,i,t]) + u_cur[b,t]
// block = (32,8): each lane owns 4 adjacent columns (float4 loads, 512B per wave per row),
// ty -> 8 partitions of the i axis (256 rows each); LDS combine across ty.
__global__ __launch_bounds__(256) void col_lse_u_kernel(const float* __restrict__ C,
                                                        const float* __restrict__ u_cur,
                                                        const float* __restrict__ v_cur,
                                                        float* __restrict__ u_next,
                                                        const int* __restrict__ done,
                                                        float log_ab) {
    int tx = threadIdx.x, ty = threadIdx.y;
    int b = blockIdx.x >> 4;                         // 16 blocks per batch
    int j0 = ((blockIdx.x & 15) << 7) + (tx << 2);   // 128 columns per block, 4 per lane
    if (*done) {                                     // frozen: copy-through
        if (ty == 0)
            *(float4*)(u_next + b * N + j0) = *(const float4*)(u_cur + b * N + j0);
        return;
    }
    const float* Cb = C + (size_t)b * N * N;
    const float* ub = u_cur + b * N;
    float4 vj = *(const float4*)(v_cur + b * N + j0);

    float m[4] = { -INFINITY, -INFINITY, -INFINITY, -INFINITY };
    float s[4] = { 0.0f, 0.0f, 0.0f, 0.0f };
    int i0 = ty * (N / 8);
    for (int t = 0; t < N / 8; ++t) {
        int i = i0 + t;
        float ui = ub[i];
        float4 c = *(const float4*)(Cb + (size_t)i * N + j0);
        lse_step(m[0], s[0], (vj.x + ui - c.x) * INV_EPS);
        lse_step(m[1], s[1], (vj.y + ui - c.y) * INV_EPS);
        lse_step(m[2], s[2], (vj.z + ui - c.z) * INV_EPS);
        lse_step(m[3], s[3], (vj.w + ui - c.w) * INV_EPS);
    }
    __shared__ float sm[8][128], ss[8][128];
    int cbase = tx << 2;
#pragma unroll
    for (int q = 0; q < 4; ++q) { sm[ty][cbase + q] = m[q]; ss[ty][cbase + q] = s[q]; }
    __syncthreads();
    if (ty == 0) {
#pragma unroll
        for (int q = 0; q < 4; ++q) {
            float mm = sm[0][cbase + q], sacc = ss[0][cbase + q];
#pragma unroll
            for (int p = 1; p < 8; ++p) lse_merge(mm, sacc, sm[p][cbase + q], ss[p][cbase + q]);
            float lse = mm + __logf(sacc);
            u_next[b * N + j0 + q] = EPS * (log_ab - lse) + ub[j0 + q];
        }
    }
}

// ---------------------------------------------------------------- v update: LSE over j (axis=-1)
// one wave per (b,i) row; lanes take float4 chunks, 4 accumulators, then shuffle-combine
__global__ __launch_bounds__(256) void row_lse_v_kernel(const float* __restrict__ C,
                                                        const float* __restrict__ u_next,
                                                        const float* __restrict__ v_cur,
                                                        float* __restrict__ v_next,
                                                        const int* __restrict__ done,
                                                        float log_ab) {
    int lane = threadIdx.x;
    int row = blockIdx.x * 8 + threadIdx.y;        // b*N + i, 0..16383
    if (*done) {
        if (lane == 0) v_next[row] = v_cur[row];
        return;
    }
    int b = row >> 11;
    const float* Crow = C + (size_t)row * N;
    const float* vb = v_cur + b * N;
    float ui = u_next[row];

    float m[4] = { -INFINITY, -INFINITY, -INFINITY, -INFINITY };
    float s[4] = { 0.0f, 0.0f, 0.0f, 0.0f };
    for (int t = 0; t < N / 128; ++t) {            // 16 steps, 128B per wave per step x4
        int j = (t * 32 + lane) * 4;
        float4 c = *(const float4*)(Crow + j);
        float4 vv = *(const float4*)(vb + j);
        lse_step(m[0], s[0], (ui + vv.x - c.x) * INV_EPS);
        lse_step(m[1], s[1], (ui + vv.y - c.y) * INV_EPS);
        lse_step(m[2], s[2], (ui + vv.z - c.z) * INV_EPS);
        lse_step(m[3], s[3], (ui + vv.w - c.w) * INV_EPS);
    }
    lse_merge(m[0], s[0], m[1], s[1]);
    lse_merge(m[2], s[2], m[3], s[3]);
    lse_merge(m[0], s[0], m[2], s[2]);
#pragma unroll
    for (int off = 16; off > 0; off >>= 1) {
        float m2 = __shfl_xor(m[0], off, 32);
        float s2 = __shfl_xor(s[0], off, 32);
        lse_merge(m[0], s[0], m2, s2);
    }
    if (lane == 0)
        v_next[row] = EPS * (log_ab - (m[0] + __logf(s[0]))) + v_cur[row];
}

// ---------------------------------------------------------------- convergence (sticky done)
__global__ __launch_bounds__(256) void diff_done_kernel(const float* u0, const float* u1,
                                                        const float* v0, const float* v1,
                                                        int* done) {
    __shared__ float red[256];
    float acc = 0.0f;
    for (int idx = threadIdx.x; idx < BATCH * N; idx += 256)
        acc += fabsf(u1[idx] - u0[idx]) + fabsf(v1[idx] - v0[idx]);
    red[threadIdx.x] = acc;
    __syncthreads();
    for (int o = 128; o > 0; o >>= 1) {
        if (threadIdx.x < o) red[threadIdx.x] += red[threadIdx.x + o];
        __syncthreads();
    }
    if (threadIdx.x == 0 && (red[0] / (float)BATCH) < THRESH) *done = 1;
}

// ---------------------------------------------------------------- pi = exp(K) in place + row sums of pi*C
__global__ __launch_bounds__(256) void pi_cost_kernel(float* __restrict__ Cpi,
                                                      const float* __restrict__ u,
                                                      const float* __restrict__ v,
                                                      float* __restrict__ rowsum) {
    int row = blockIdx.x;                 // b*N + i
    int b = row >> 11;
    float* Crow = Cpi + (size_t)row * N;
    const float* vb = v + b * N;
    float ui = u[row];
    float acc = 0.0f;
    for (int t = 0; t < N / 1024; ++t) {  // 2 steps of 256 threads x float4
        int j = (t * 256 + threadIdx.x) * 4;
        float4 c = *(const float4*)(Crow + j);
        float4 vv = *(const float4*)(vb + j);
        float4 p;
        p.x = expf((ui + vv.x - c.x) * INV_EPS);
        p.y = expf((ui + vv.y - c.y) * INV_EPS);
        p.z = expf((ui + vv.z - c.z) * INV_EPS);
        p.w = expf((ui + vv.w - c.w) * INV_EPS);
        *(float4*)(Crow + j) = p;         // overwrite C with pi (one-to-one, safe)
        acc += p.x * c.x + p.y * c.y + p.z * c.z + p.w * c.w;
    }
    __shared__ float red[256];
    red[threadIdx.x] = acc;
    __syncthreads();
    for (int o = 128; o > 0; o >>= 1) {
        if (threadIdx.x < o) red[threadIdx.x] += red[threadIdx.x + o];
        __syncthreads();
    }
    if (threadIdx.x == 0) rowsum[row] = red[0];
}

__global__ __launch_bounds__(256) void cost_reduce_kernel(const float* __restrict__ rowsum,
                                                          float* __restrict__ cost) {
    int b = blockIdx.x;
    float acc = 0.0f;
    for (int i = threadIdx.x; i < N; i += 256) acc += rowsum[b * N + i];
    __shared__ float red[256];
    red[threadIdx.x] = acc;
    __syncthreads();
    for (int o = 128; o > 0; o >>= 1) {
        if (threadIdx.x < o) red[threadIdx.x] += red[threadIdx.x + o];
        __syncthreads();
    }
    if (threadIdx.x == 0) cost[b] = red[0];
}

// ---------------------------------------------------------------- host
extern "C" void kernel_launch(void* const* d_in, const int* in_sizes, int n_in,
                              void* d_out, int out_size, void* d_ws, size_t ws_size,
                              hipStream_t stream) {
    const float* x = (const float*)d_in[0];
    const float* y = (const float*)d_in[1];
    float* cost = (float*)d_out;            // [8]
    float* Cpi  = (float*)d_out + 8;        // [8,2048,2048]: C during iterations, pi at the end

    float* ws     = (float*)d_ws;
    float* x2     = ws;                     // 16384
    float* y2     = ws + 16384;             // 16384
    float* ubuf[2] = { ws + 32768, ws + 49152 };
    float* vbuf[2] = { ws + 65536, ws + 81920 };
    float* rowsum = ws + 98304;             // 16384
    int*   done   = (int*)(ws + 114688);

    const float log_ab = logf(1.0f / (float)N + 1e-8f);

    init_kernel<<<64, 256, 0, stream>>>(ubuf[0], vbuf[0], done);
    norms_kernel<<<64, 256, 0, stream>>>(x, y, x2, y2);
    gemm_cost_kernel<<<4096, dim3(32, 8), 0, stream>>>(x, y, x2, y2, Cpi);

    int cur = 0;
    for (int it = 0; it < 100; ++it) {
        int nxt = cur ^ 1;
        col_lse_u_kernel<<<128, dim3(32, 8), 0, stream>>>(Cpi, ubuf[cur], vbuf[cur], ubuf[nxt], done, log_ab);
        row_lse_v_kernel<<<2048, dim3(32, 8), 0, stream>>>(Cpi, ubuf[nxt], vbuf[cur], vbuf[nxt], done, log_ab);
        diff_done_kernel<<<1, 256, 0, stream>>>(ubuf[cur], ubuf[nxt], vbuf[cur], vbuf[nxt], done);
        cur = nxt;
    }

    pi_cost_kernel<<<BATCH * N, 256, 0, stream>>>(Cpi, ubuf[cur], vbuf[cur], rowsum);
    cost_reduce_kernel<<<BATCH, 256, 0, stream>>>(rowsum, cost);
}